// Shrinkagev3ppp1_26860725469560
// MI455X (gfx1250) — compile-verified
//
#include <hip/hip_runtime.h>
#include <hip/hip_bf16.h>

// Fixed problem dims from setup_inputs()
#define B_   8
#define C_   128
#define MIP_ 32
#define L_   65536
#define ROWS_ (B_ * C_)          // 1024 (b*C + c)
#define L4_  (L_ / 4)            // 16384 float4 per row
#define BN_EPS_ 1e-5f

typedef float v2f __attribute__((ext_vector_type(2)));
typedef float v4f __attribute__((ext_vector_type(4)));
typedef float v8f __attribute__((ext_vector_type(8)));

// ---------------------------------------------------------------------------
// Kernel 1: per-(b,c) mean over L.  1024 blocks x 256 threads, float4 streams.
// Memory bound: 268 MB read @ 23.3 TB/s ~ 11.5 us.
// ---------------------------------------------------------------------------
__global__ __launch_bounds__(256)
void row_mean_kernel(const float* __restrict__ x, float* __restrict__ avg) {
    const int row = blockIdx.x;
    const v4f* xp = (const v4f*)(x + (size_t)row * L_);
    v4f acc = {};
    for (int i = threadIdx.x; i < L4_; i += 256) {
        v4f v = xp[i];
        acc.x += v.x; acc.y += v.y; acc.z += v.z; acc.w += v.w;
    }
    __shared__ float red[256];
    red[threadIdx.x] = (acc.x + acc.y) + (acc.z + acc.w);
    __syncthreads();
    for (int off = 128; off > 0; off >>= 1) {
        if (threadIdx.x < off) red[threadIdx.x] += red[threadIdx.x + off];
        __syncthreads();
    }
    if (threadIdx.x == 0) avg[row] = red[0] * (1.0f / (float)L_);
}

// ---------------------------------------------------------------------------
// Kernel 2: SE MLP on the matrix core, one wave (wave32), EXEC all ones.
//   H(32x16)  = w1(32x128)  x avgT(128x16-padded)   -> BN-fold + ReLU
//   G(128x16) = w2(128x32)  x Hr(32x16)             -> hardsigmoid
//   thres[b*C+c] = avg[b*C+c] * G[c][b]
// Uses V_WMMA_F32_16X16X4_F32; batch dim b padded 8->16 with zero columns.
// ---------------------------------------------------------------------------
__global__ __launch_bounds__(32)
void se_mlp_wmma_kernel(const float* __restrict__ avg,
                        const float* __restrict__ w1, const float* __restrict__ b1,
                        const float* __restrict__ gam, const float* __restrict__ bet,
                        const float* __restrict__ mean, const float* __restrict__ var,
                        const float* __restrict__ w2, const float* __restrict__ b2,
                        float* __restrict__ thres) {
    __shared__ float avg_lds[ROWS_];       // [b*C + c]
    __shared__ float s_lds[MIP_];          // folded BN scale
    __shared__ float t_lds[MIP_];          // folded BN shift (incl. conv bias)
    __shared__ float h_lds[MIP_ * 16];     // Hr in [k][n] layout for stage-2 B

    const int l  = threadIdx.x;
    const int lo = l & 15;                 // A-row / B,C,D column index
    const int hi = l >> 4;                 // K / M sub-split

    for (int i = l; i < ROWS_; i += 32) avg_lds[i] = avg[i];
    {   // fold conv bias + BN into  h = raw * s + t
        float s = gam[l] * rsqrtf(var[l] + BN_EPS_);
        s_lds[l] = s;
        t_lds[l] = (b1[l] - mean[l]) * s + bet[l];
    }
    __syncthreads();

    // ---- Stage 1:  H = w1 x avgT,  2 M-tiles, K=128 in 32 steps of 4 ----
    const float colmask = (lo < 8) ? 1.0f : 0.0f;   // zero-pad batch columns
    const int   bsel    = (lo & 7) * C_;
    v8f acc0 = {}; v8f acc1 = {};
    for (int kk = 0; kk < 32; ++kk) {
        const int k0 = 4 * kk + 2 * hi;
        v2f a0, a1, b;
        a0.x = w1[lo * C_ + k0];          a0.y = w1[lo * C_ + k0 + 1];
        a1.x = w1[(16 + lo) * C_ + k0];   a1.y = w1[(16 + lo) * C_ + k0 + 1];
        b.x  = avg_lds[bsel + k0]     * colmask;
        b.y  = avg_lds[bsel + k0 + 1] * colmask;
        acc0 = __builtin_amdgcn_wmma_f32_16x16x4_f32(false, a0, false, b,
                                                     (short)0, acc0, false, false);
        acc1 = __builtin_amdgcn_wmma_f32_16x16x4_f32(false, a1, false, b,
                                                     (short)0, acc1, false, false);
    }
    // BN + ReLU; restage D-layout -> B-operand layout h_lds[k*16 + n]
#pragma unroll
    for (int v = 0; v < 8; ++v) {
        const int m0 = v + 8 * hi;        // tile 0: m = 0..15
        const int m1 = m0 + 16;           // tile 1: m = 16..31
        h_lds[m0 * 16 + lo] = fmaxf(acc0[v] * s_lds[m0] + t_lds[m0], 0.0f);
        h_lds[m1 * 16 + lo] = fmaxf(acc1[v] * s_lds[m1] + t_lds[m1], 0.0f);
    }
    __syncthreads();

    // ---- Stage 2:  G = w2 x Hr,  8 M-tiles of 16 channels, K=32 ----
    for (int ct = 0; ct < 8; ++ct) {
        v8f acc = {};
        for (int kk = 0; kk < 8; ++kk) {
            const int k0 = 4 * kk + 2 * hi;
            v2f a, b;
            a.x = w2[(ct * 16 + lo) * MIP_ + k0];
            a.y = w2[(ct * 16 + lo) * MIP_ + k0 + 1];
            b.x = h_lds[k0 * 16 + lo];
            b.y = h_lds[(k0 + 1) * 16 + lo];
            acc = __builtin_amdgcn_wmma_f32_16x16x4_f32(false, a, false, b,
                                                        (short)0, acc, false, false);
        }
        if (lo < 8) {                      // valid batch columns only
#pragma unroll
            for (int v = 0; v < 8; ++v) {
                const int c = ct * 16 + v + 8 * hi;
                float g = acc[v] + b2[c];
                g = fminf(fmaxf((g + 3.0f) * (1.0f / 6.0f), 0.0f), 1.0f); // hardsigmoid
                thres[lo * C_ + c] = avg_lds[lo * C_ + c] * g;
            }
        }
    }
}

// ---------------------------------------------------------------------------
// Kernel 3: elementwise soft-threshold shrinkage.  float4 streams, NT stores.
// 268 MB in + 268 MB out ~ 23 us at HBM roof.
// ---------------------------------------------------------------------------
__device__ __forceinline__ float shrink1(float x, float t, float oma) {
    float sub = fmaxf(fabsf(x) - t, 0.0f);
    float val = sub + oma * t;
    float o   = (sub > 0.0f) ? val : 0.0f;
    return (x > 0.0f) ? o : ((x < 0.0f) ? -o : 0.0f);   // sign(0) == 0
}

__global__ __launch_bounds__(256)
void shrink_kernel(const v4f* __restrict__ x4, const float* __restrict__ thres,
                   const float* __restrict__ a, v4f* __restrict__ out4) {
    const size_t i   = (size_t)blockIdx.x * 256 + threadIdx.x;   // float4 index
    const int    row = (int)(i >> 14);                           // / (L/4)
    const float  t   = thres[row];                               // L2-resident (4 KB)
    const float  oma = 1.0f - a[0];
    v4f xv = x4[i];
    v4f ov;
    ov.x = shrink1(xv.x, t, oma);
    ov.y = shrink1(xv.y, t, oma);
    ov.z = shrink1(xv.z, t, oma);
    ov.w = shrink1(xv.w, t, oma);
    __builtin_nontemporal_store(ov, &out4[i]);   // don't evict x from L2
}

// ---------------------------------------------------------------------------
extern "C" void kernel_launch(void* const* d_in, const int* in_sizes, int n_in,
                              void* d_out, int out_size, void* d_ws, size_t ws_size,
                              hipStream_t stream) {
    const float* x    = (const float*)d_in[0];
    const float* a    = (const float*)d_in[1];
    const float* w1   = (const float*)d_in[2];
    const float* b1   = (const float*)d_in[3];
    const float* gam  = (const float*)d_in[4];
    const float* bet  = (const float*)d_in[5];
    const float* mean = (const float*)d_in[6];
    const float* var  = (const float*)d_in[7];
    const float* w2   = (const float*)d_in[8];
    const float* b2   = (const float*)d_in[9];

    float* ws    = (float*)d_ws;
    float* avg   = ws;            // 1024 floats
    float* thres = ws + ROWS_;    // 1024 floats

    row_mean_kernel<<<ROWS_, 256, 0, stream>>>(x, avg);
    se_mlp_wmma_kernel<<<1, 32, 0, stream>>>(avg, w1, b1, gam, bet, mean, var,
                                             w2, b2, thres);
    const int n4 = (B_ * C_ * L_) / 4;            // 16,777,216
    shrink_kernel<<<n4 / 256, 256, 0, stream>>>((const v4f*)x, thres, a,
                                                (v4f*)d_out);
}